// MultiCodeDecoderWrapper_65884798320664
// MI455X (gfx1250) — compile-verified
//
#include <hip/hip_runtime.h>
#include <hip/hip_bf16.h>
#include <math.h>

// ---------------------------------------------------------------------------
// Model constants (from reference)
// ---------------------------------------------------------------------------
#define DMODEL 1024
#define NHEADS 16
#define NKV    4
#define HDIM   64
#define KVD    (NKV * HDIM)     // 256
#define NLAYER 6
#define FFN    4096
#define VOCAB  4096
#define NCODE  15
#define SLEN   16
#define EPSV   1e-6f

typedef float v2f __attribute__((ext_vector_type(2)));
typedef float v8f __attribute__((ext_vector_type(8)));
typedef int   v4i __attribute__((ext_vector_type(4)));

#define AS1 __attribute__((address_space(1)))
#define AS3 __attribute__((address_space(3)))

#if __has_builtin(__builtin_amdgcn_global_load_async_to_lds_b128)
#define HAVE_ASYNC 1
#else
#define HAVE_ASYNC 0
#endif

#if HAVE_ASYNC
#if __has_builtin(__builtin_amdgcn_s_wait_asynccnt)
#define WAIT_ASYNC(n) __builtin_amdgcn_s_wait_asynccnt(n)
#else
#define WAIT_ASYNC(n) asm volatile("s_wait_asynccnt %0" ::"n"(n) : "memory")
#endif
#define WAIT_DS0() asm volatile("s_wait_dscnt 0x0" ::: "memory")
#else
#define WAIT_ASYNC(n) ((void)0)
#define WAIT_DS0() ((void)0)
#endif

// ---------------------------------------------------------------------------
// setup: cos/sin for RoPE (pos = cache_length), copy input embeds -> hidden
// ---------------------------------------------------------------------------
__global__ void setup_kernel(const float* __restrict__ emb,
                             const int* __restrict__ cache_length,
                             const float* __restrict__ inv_freq,
                             float* __restrict__ cosv, float* __restrict__ sinv,
                             float* __restrict__ hidden) {
    int t = threadIdx.x;
    if (t < HDIM / 2) {
        float pos = (float)cache_length[0];
        float f = inv_freq[t] * pos;
        float c = cosf(f), s = sinf(f);
        cosv[t] = c; cosv[t + HDIM / 2] = c;
        sinv[t] = s; sinv[t + HDIM / 2] = s;
    }
    for (int i = t; i < DMODEL; i += blockDim.x) hidden[i] = emb[i];
}

// ---------------------------------------------------------------------------
// RMSNorm over n elements (one block)
// ---------------------------------------------------------------------------
__global__ void rmsnorm_kernel(const float* __restrict__ x,
                               const float* __restrict__ w,
                               float* __restrict__ out, int n) {
    __shared__ float red[32];
    int tid = threadIdx.x;
    float s = 0.0f;
    for (int i = tid; i < n; i += blockDim.x) { float v = x[i]; s += v * v; }
    for (int off = 16; off > 0; off >>= 1) s += __shfl_down(s, off, 32);
    if ((tid & 31) == 0) red[tid >> 5] = s;
    __syncthreads();
    int nw = blockDim.x >> 5;
    if (tid < 32) {
        float t2 = (tid < nw) ? red[tid] : 0.0f;
        for (int off = 16; off > 0; off >>= 1) t2 += __shfl_down(t2, off, 32);
        if (tid == 0) red[0] = t2;
    }
    __syncthreads();
    float scale = rsqrtf(red[0] / (float)n + EPSV);
    for (int i = tid; i < n; i += blockDim.x) out[i] = x[i] * scale * w[i];
}

// ---------------------------------------------------------------------------
// WMMA GEMV with async global->LDS weight staging.
//
//   y[row] = sum_k W[row,k] * x[k]   (+ addsrc[row] if non-null)
//
// One wave computes 16 rows. K is processed in 64-wide tiles:
//   - tile (16 rows x 64 cols fp32 = 4KB) staged into LDS with 8
//     GLOBAL_LOAD_ASYNC_TO_LDS_B128 ops (each a fully coalesced 512B burst:
//     lanes 0-15 cover row 2i, lanes 16-31 cover row 2i+1), double-buffered
//     and retired with s_wait_asynccnt.
//   - 16 x V_WMMA_F32_16X16X4_F32 per tile. A fragment (16x4 f32: lanes 0-15
//     hold K=0,1; lanes 16-31 hold K=2,3; M = lane&15) read from LDS
//     (row stride 68 words -> conflict-free banks). B = x-chunk broadcast into
//     every column, so every column of D equals W_tile @ x_chunk.
//   - C/D 16x16: VGPR r: lanes 0-15 -> M=r, lanes 16-31 -> M=8+r.
//
// Requirements: M % 16 == 0, K % 64 == 0. Launch block = 128 (4 waves),
// 64 rows per block.
// ---------------------------------------------------------------------------
#define TS   68                 // LDS tile row stride in floats
#define GWPB 4                  // waves per block

__device__ __forceinline__ void stage_tile(const float* Wb, size_t K,
                                           float* lds, int lane) {
    int half = lane >> 4;              // 0 or 1
    int c4   = (lane & 15) * 4;        // float offset within a row (16B units)
#pragma unroll
    for (int i = 0; i < 8; ++i) {
        int row = 2 * i + half;
        const float* g = Wb + (size_t)row * K + c4;
        float* l = lds + row * TS + c4;
#if HAVE_ASYNC
        __builtin_amdgcn_global_load_async_to_lds_b128(
            (AS1 v4i*)(AS1 void*)(float*)g, (AS3 v4i*)(AS3 void*)l, 0, 0);
#else
        float4 v = *(const float4*)g;
        *(float4*)l = v;
#endif
    }
}

__global__ void gemv_wmma_kernel(const float* __restrict__ W,
                                 const float* __restrict__ x,
                                 float* __restrict__ y,
                                 const float* __restrict__ addsrc,
                                 int M, int K) {
    __shared__ __align__(16) float tile[GWPB][2][16 * TS];
    int lane = threadIdx.x & 31;
    int wave = threadIdx.x >> 5;
    int rowBase = (blockIdx.x * GWPB + wave) * 16;
    if (rowBase >= M) return;            // wave-uniform: EXEC stays all-1s

    int r    = lane & 15;
    int ksel = (lane >> 4) << 1;         // 0 for lanes 0-15, 2 for 16-31
    const float* Wb = W + (size_t)rowBase * (size_t)K;
    float* t0 = &tile[wave][0][0];
    float* t1 = &tile[wave][1][0];

    int numTiles = K >> 6;
    stage_tile(Wb, K, t0, lane);

    v8f c = {};
    for (int t = 0; t < numTiles; ++t) {
        float* tb = (t & 1) ? t1 : t0;
        if (t + 1 < numTiles) {
            WAIT_DS0();                  // prior reads of other buffer done
            stage_tile(Wb + (size_t)(t + 1) * 64, K, (t & 1) ? t0 : t1, lane);
            WAIT_ASYNC(8);               // tile t's 8 async ops retired
        } else {
            WAIT_ASYNC(0);
        }
        int kt = t << 6;
#pragma unroll
        for (int j = 0; j < 16; ++j) {
            int o = 4 * j + ksel;
            v2f a, b;
            a.x = tb[r * TS + o];
            a.y = tb[r * TS + o + 1];
            b.x = x[kt + o];
            b.y = x[kt + o + 1];
            c = __builtin_amdgcn_wmma_f32_16x16x4_f32(false, a, false, b,
                                                      (short)0, c, false, false);
        }
    }
    // every column identical; lane 0 holds y[rowBase+0..7], lane 16 holds +8..15
    if ((lane & 15) == 0) {
        int mbase = rowBase + ((lane >> 4) << 3);
#pragma unroll
        for (int q = 0; q < 8; ++q) {
            float val = c[q];
            if (addsrc) val += addsrc[mbase + q];
            y[mbase + q] = val;
        }
    }
}

// ---------------------------------------------------------------------------
// Per-head RMSNorm (qn/kn) + RoPE on q (16 heads) and k (4 heads).
// grid = NHEADS + NKV blocks, 64 threads each.
// ---------------------------------------------------------------------------
__global__ void qknorm_rope_kernel(float* __restrict__ q, float* __restrict__ k,
                                   const float* __restrict__ qn,
                                   const float* __restrict__ kn,
                                   const float* __restrict__ cosv,
                                   const float* __restrict__ sinv) {
    int head = blockIdx.x;
    float* vec; const float* w;
    if (head < NHEADS) { vec = q + head * HDIM; w = qn; }
    else               { vec = k + (head - NHEADS) * HDIM; w = kn; }

    int t = threadIdx.x;                // 0..63
    float v = vec[t];
    __shared__ float red[2];
    float s = v * v;
    for (int off = 16; off > 0; off >>= 1) s += __shfl_down(s, off, 32);
    if ((t & 31) == 0) red[t >> 5] = s;
    __syncthreads();
    float scale = rsqrtf((red[0] + red[1]) / (float)HDIM + EPSV);
    float nv = v * scale * w[t];

    __shared__ float nh[HDIM];
    nh[t] = nv;
    __syncthreads();
    float rot = (t < HDIM / 2) ? -nh[t + HDIM / 2] : nh[t - HDIM / 2];
    vec[t] = nv * cosv[t] + rot * sinv[t];
}

// ---------------------------------------------------------------------------
// Update one layer's slice of the output K/V caches with umask-blended column.
// nkc_l / nvc_l: (KVD, S) row-major. 2*KVD*S threads.
// ---------------------------------------------------------------------------
__global__ void cache_update_kernel(float* __restrict__ nkc_l,
                                    float* __restrict__ nvc_l,
                                    const float* __restrict__ k,
                                    const float* __restrict__ v,
                                    const float* __restrict__ umask) {
    int idx = blockIdx.x * blockDim.x + threadIdx.x;
    const int half = KVD * SLEN;
    if (idx < half) {
        int d = idx >> 4, s = idx & (SLEN - 1);
        float m = umask[s];
        nkc_l[idx] = nkc_l[idx] * (1.0f - m) + k[d] * m;
    } else if (idx < 2 * half) {
        int j = idx - half;
        int d = j >> 4, s = j & (SLEN - 1);
        float m = umask[s];
        nvc_l[j] = nvc_l[j] * (1.0f - m) + v[d] * m;
    }
}

// ---------------------------------------------------------------------------
// Attention for one layer. grid = NHEADS blocks of 64 threads.
// kc/vc: this layer's (KVD, S) updated cache. GQA: head h uses kv group h/4.
// ---------------------------------------------------------------------------
__global__ void attn_kernel(const float* __restrict__ q,
                            const float* __restrict__ kc,
                            const float* __restrict__ vc,
                            const float* __restrict__ pmask,
                            float* __restrict__ ao) {
    int h = blockIdx.x;
    int g = h >> 2;                     // n_rep = NHEADS/NKV = 4
    int t = threadIdx.x;
    __shared__ float qs[HDIM];
    __shared__ float aw[SLEN];
    qs[t] = q[h * HDIM + t];
    __syncthreads();
    if (t < SLEN) {
        float sc = 0.0f;
        const float* kcol = kc + g * HDIM * SLEN + t;   // k[g][d][t], stride S
        for (int d = 0; d < HDIM; ++d) sc += qs[d] * kcol[d * SLEN];
        aw[t] = sc * 0.125f + pmask[t];                 // 1/sqrt(64)
    }
    __syncthreads();
    if (t == 0) {
        float mx = aw[0];
        for (int s = 1; s < SLEN; ++s) mx = fmaxf(mx, aw[s]);
        float sum = 0.0f;
        for (int s = 0; s < SLEN; ++s) { float e = expf(aw[s] - mx); aw[s] = e; sum += e; }
        float inv = 1.0f / sum;
        for (int s = 0; s < SLEN; ++s) aw[s] *= inv;
    }
    __syncthreads();
    float acc = 0.0f;
    const float* vrow = vc + (g * HDIM + t) * SLEN;
    for (int s = 0; s < SLEN; ++s) acc += aw[s] * vrow[s];
    ao[h * HDIM + t] = acc;
}

// ---------------------------------------------------------------------------
// out = silu(gate) * up
// ---------------------------------------------------------------------------
__global__ void silu_mul_kernel(const float* __restrict__ g,
                                const float* __restrict__ u,
                                float* __restrict__ o, int n) {
    int i = blockIdx.x * blockDim.x + threadIdx.x;
    if (i < n) {
        float x = g[i];
        o[i] = (x / (1.0f + expf(-x))) * u[i];
    }
}

// ---------------------------------------------------------------------------
// Host-side orchestration
// ---------------------------------------------------------------------------
extern "C" void kernel_launch(void* const* d_in, const int* in_sizes, int n_in,
                              void* d_out, int out_size, void* d_ws, size_t ws_size,
                              hipStream_t stream) {
    (void)in_sizes; (void)n_in; (void)out_size; (void)ws_size;

    const float* input_embeds = (const float*)d_in[0];
    const int*   cache_length = (const int*)  d_in[1];
    const float* key_cache    = (const float*)d_in[2];
    const float* pmask        = (const float*)d_in[3];
    const float* umask        = (const float*)d_in[4];
    const float* value_cache  = (const float*)d_in[5];
    const float* inv_freq     = (const float*)d_in[6];
    const float* Wq           = (const float*)d_in[7];
    const float* Wk           = (const float*)d_in[8];
    const float* Wv           = (const float*)d_in[9];
    const float* Wo           = (const float*)d_in[10];
    const float* ln1_w        = (const float*)d_in[11];
    const float* ln2_w        = (const float*)d_in[12];
    const float* qn_w         = (const float*)d_in[13];
    const float* kn_w         = (const float*)d_in[14];
    const float* Wgate        = (const float*)d_in[15];
    const float* Wup          = (const float*)d_in[16];
    const float* Wdown        = (const float*)d_in[17];
    const float* norm_w       = (const float*)d_in[18];
    const float* lm_head_w    = (const float*)d_in[19];

    // outputs, concatenated flat in return order
    float* out        = (float*)d_out;
    float* logits     = out;                            // NCODE*VOCAB = 61440
    float* hidden_out = out + NCODE * VOCAB;            // 1024
    float* nkc        = hidden_out + DMODEL;            // L*KVD*S = 24576
    float* nvc        = nkc + NLAYER * KVD * SLEN;      // 24576

    // workspace
    float* ws      = (float*)d_ws;
    float* hidden  = ws;                 // 1024
    float* hbuf    = ws + 1024;          // 1024
    float* qbuf    = ws + 2048;          // 1024
    float* kbuf    = ws + 3072;          // 256
    float* vbuf    = ws + 3328;          // 256
    float* aobuf   = ws + 3584;          // 1024
    float* gatebuf = ws + 4608;          // 4096
    float* upbuf   = ws + 8704;          // 4096
    float* mulbuf  = ws + 12800;         // 4096
    float* cosbuf  = ws + 16896;         // 64
    float* sinbuf  = ws + 16960;         // 64

    // 1) bulk-copy caches into output; per-layer column blended later
    (void)hipMemcpyAsync(nkc, key_cache,
                         (size_t)NLAYER * KVD * SLEN * sizeof(float),
                         hipMemcpyDeviceToDevice, stream);
    (void)hipMemcpyAsync(nvc, value_cache,
                         (size_t)NLAYER * KVD * SLEN * sizeof(float),
                         hipMemcpyDeviceToDevice, stream);

    // 2) RoPE tables + hidden init
    setup_kernel<<<1, 256, 0, stream>>>(input_embeds, cache_length, inv_freq,
                                        cosbuf, sinbuf, hidden);

    // gemv launch: 128 threads = 4 waves = 64 rows per block
    const int GB = 32 * GWPB;
    const int RPB = 16 * GWPB;

    for (int l = 0; l < NLAYER; ++l) {
        const float* Wq_ = Wq + (size_t)l * NHEADS * HDIM * DMODEL;
        const float* Wk_ = Wk + (size_t)l * KVD * DMODEL;
        const float* Wv_ = Wv + (size_t)l * KVD * DMODEL;
        const float* Wo_ = Wo + (size_t)l * DMODEL * NHEADS * HDIM;
        const float* Wg_ = Wgate + (size_t)l * FFN * DMODEL;
        const float* Wu_ = Wup   + (size_t)l * FFN * DMODEL;
        const float* Wd_ = Wdown + (size_t)l * DMODEL * FFN;
        float* nkc_l = nkc + (size_t)l * KVD * SLEN;
        float* nvc_l = nvc + (size_t)l * KVD * SLEN;

        // attention block
        rmsnorm_kernel<<<1, 256, 0, stream>>>(hidden, ln1_w + l * DMODEL, hbuf, DMODEL);
        gemv_wmma_kernel<<<DMODEL / RPB, GB, 0, stream>>>(Wq_, hbuf, qbuf, nullptr, DMODEL, DMODEL);
        gemv_wmma_kernel<<<KVD    / RPB, GB, 0, stream>>>(Wk_, hbuf, kbuf, nullptr, KVD, DMODEL);
        gemv_wmma_kernel<<<KVD    / RPB, GB, 0, stream>>>(Wv_, hbuf, vbuf, nullptr, KVD, DMODEL);
        qknorm_rope_kernel<<<NHEADS + NKV, HDIM, 0, stream>>>(qbuf, kbuf,
                                                              qn_w + l * HDIM, kn_w + l * HDIM,
                                                              cosbuf, sinbuf);
        cache_update_kernel<<<(2 * KVD * SLEN + 255) / 256, 256, 0, stream>>>(
            nkc_l, nvc_l, kbuf, vbuf, umask);
        attn_kernel<<<NHEADS, HDIM, 0, stream>>>(qbuf, nkc_l, nvc_l, pmask, aobuf);
        gemv_wmma_kernel<<<DMODEL / RPB, GB, 0, stream>>>(Wo_, aobuf, hidden, hidden, DMODEL, DMODEL);

        // MLP block
        rmsnorm_kernel<<<1, 256, 0, stream>>>(hidden, ln2_w + l * DMODEL, hbuf, DMODEL);
        gemv_wmma_kernel<<<FFN / RPB, GB, 0, stream>>>(Wg_, hbuf, gatebuf, nullptr, FFN, DMODEL);
        gemv_wmma_kernel<<<FFN / RPB, GB, 0, stream>>>(Wu_, hbuf, upbuf,   nullptr, FFN, DMODEL);
        silu_mul_kernel<<<(FFN + 255) / 256, 256, 0, stream>>>(gatebuf, upbuf, mulbuf, FFN);
        gemv_wmma_kernel<<<DMODEL / RPB, GB, 0, stream>>>(Wd_, mulbuf, hidden, hidden, DMODEL, FFN);
    }

    // final norm -> hidden_out output slot, then lm_head GEMV (15 heads fused)
    rmsnorm_kernel<<<1, 256, 0, stream>>>(hidden, norm_w, hidden_out, DMODEL);
    gemv_wmma_kernel<<<(NCODE * VOCAB) / RPB, GB, 0, stream>>>(
        lm_head_w, hidden_out, logits, nullptr, NCODE * VOCAB, DMODEL);
}